// GCN_73890617360513
// MI455X (gfx1250) — compile-verified
//
#include <hip/hip_runtime.h>
#include <math.h>

// ---------------------------------------------------------------------------
// GCN, 2-layer, DGL GraphConv norm='both'.
//   L1: h = relu( Dd^-1/2 * A * Ds^-1/2 * (X @ W1) + b1 )
//   L2: out =     Dd^-1/2 * A * Ds^-1/2 * (h @ W2) + b2
// GEMMs use V_WMMA_F32_16X16X4_F32 (full fp32 — matches reference precision).
// W1/W2 are pre-swizzled into per-lane WMMA B-fragment order so the hot loop
// does one coalesced b64 load per fragment (batched in s_clause) instead of
// two strided b32 loads with a full-latency wait before every WMMA.
// Aggregation is edge-parallel global_atomic_add_f32 (bandwidth bound:
// ~2 GB traffic @ 23.3 TB/s dominates; GEMM FLOPs are negligible).
// ---------------------------------------------------------------------------

typedef __attribute__((ext_vector_type(2))) float v2f;
typedef __attribute__((ext_vector_type(8))) float v8f;

#define GCN_NODES   50000
#define GCN_INF     512
#define GCN_HID     128
#define GCN_CLS     40

#define GCN_KS1     (GCN_INF / 4)    // 128 K-steps, layer 1
#define GCN_KS2     (GCN_HID / 4)    // 32  K-steps, layer 2

static __device__ __forceinline__ void atomAddF(float* p, float v) {
  unsafeAtomicAdd(p, v);   // lowers to global_atomic_add_f32 on gfx1250
}

// ---- degree accumulation --------------------------------------------------
__global__ void gcn_deg_kernel(const int* __restrict__ src, const int* __restrict__ dst,
                               float* __restrict__ deg_out, float* __restrict__ deg_in,
                               int n_edges) {
  int e = blockIdx.x * blockDim.x + threadIdx.x;
  if (e < n_edges) {
    atomAddF(&deg_out[src[e]], 1.0f);
    atomAddF(&deg_in[dst[e]], 1.0f);
  }
}

// ---- norms ----------------------------------------------------------------
__global__ void gcn_norm_kernel(const float* __restrict__ deg_out,
                                const float* __restrict__ deg_in,
                                float* __restrict__ norm_src,
                                float* __restrict__ norm_dst, int n) {
  int i = blockIdx.x * blockDim.x + threadIdx.x;
  if (i < n) {
    norm_src[i] = rsqrtf(fmaxf(deg_out[i], 1.0f));
    norm_dst[i] = rsqrtf(fmaxf(deg_in[i], 1.0f));
  }
}

// ---- pack W1 into B-fragment order ----------------------------------------
// W1S layout: [128 ksteps][8 ntiles][32 lanes] x float2 (256 KB, L2 resident).
// B fragment (4x16 f32, ISA 7.12.2): lanes 0-15 -> K=kk,kk+1 ; lanes 16-31 ->
// K=kk+2,kk+3 ; N = lane&15 within the tile.
__global__ void gcn_pack_w1(const float* __restrict__ W1, float* __restrict__ W1S) {
  int idx = blockIdx.x * blockDim.x + threadIdx.x;    // 128*8*32 = 32768
  if (idx >= GCN_KS1 * 8 * 32) return;
  int lane = idx & 31;
  int nt   = (idx >> 5) & 7;
  int s    = idx >> 8;
  int lm = lane & 15, kh = lane >> 4;
  int kk = 4 * s + kh * 2;
  int n  = nt * 16 + lm;
  W1S[idx * 2 + 0] = W1[(size_t)(kk + 0) * GCN_HID + n];
  W1S[idx * 2 + 1] = W1[(size_t)(kk + 1) * GCN_HID + n];
}

// ---- pack W2 into B-fragment order (zero-padded N: 40 -> 48) --------------
// W2S layout: [32 ksteps][3 ntiles][32 lanes] x float2 (24 KB).
__global__ void gcn_pack_w2(const float* __restrict__ W2, float* __restrict__ W2S) {
  int idx = blockIdx.x * blockDim.x + threadIdx.x;    // 32*3*32 = 3072
  if (idx >= GCN_KS2 * 3 * 32) return;
  int lane = idx & 31;
  int t    = idx >> 5;          // t = s*3 + nt
  int nt = t % 3, s = t / 3;
  int lm = lane & 15, kh = lane >> 4;
  int kk = 4 * s + kh * 2;
  int n  = nt * 16 + lm;
  float v0 = 0.0f, v1 = 0.0f;
  if (n < GCN_CLS) {
    v0 = W2[(size_t)(kk + 0) * GCN_CLS + n];
    v1 = W2[(size_t)(kk + 1) * GCN_CLS + n];
  }
  W2S[idx * 2 + 0] = v0;
  W2S[idx * 2 + 1] = v1;
}

// ---- layer-1 GEMM: H1 = (X @ W1) * norm_src[row] --------------------------
// One wave per 16-row M tile; the wave covers all 8 N-tiles (128 cols) so X
// is streamed exactly once. All 8 B fragments are fetched into a register
// array first (one s_clause of coalesced b64 loads), then 8 WMMAs issue
// back-to-back.
__global__ void __launch_bounds__(32) gcn_gemm1_wmma(
    const float* __restrict__ X, const float* __restrict__ W1S,
    const float* __restrict__ norm_src, float* __restrict__ H1) {
  const int lane = threadIdx.x & 31;
  const int lm   = lane & 15;
  const int kh   = lane >> 4;              // 0 or 1: which K half-pair
  const int m0   = blockIdx.x * 16;
  const int row  = m0 + lm;

  v8f acc[8] = {};
  const float* xrow = X + (size_t)row * GCN_INF;
  const v2f*   Bv   = (const v2f*)W1S;

#pragma unroll 2
  for (int s = 0; s < GCN_KS1; ++s) {
    v2f a = *(const v2f*)(xrow + 4 * s + kh * 2);   // global_load_b64
    v2f bf[8];
#pragma unroll
    for (int nt = 0; nt < 8; ++nt)
      bf[nt] = Bv[(s * 8 + nt) * 32 + lane];        // coalesced b64 loads
#pragma unroll
    for (int nt = 0; nt < 8; ++nt)
      acc[nt] = __builtin_amdgcn_wmma_f32_16x16x4_f32(
          false, a, false, bf[nt], (short)0, acc[nt], false, false);
  }

#pragma unroll
  for (int r = 0; r < 8; ++r) {
    const int mrow = m0 + kh * 8 + r;               // C layout: VGPR r -> M=r / r+8
    const float ns = norm_src[mrow];
#pragma unroll
    for (int nt = 0; nt < 8; ++nt) {
      H1[(size_t)mrow * GCN_HID + nt * 16 + lm] = acc[nt][r] * ns;
    }
  }
}

// ---- layer-1 scatter: AGG1[dst] += H1[src], 128 feats ---------------------
// One wave per edge: each lane does one float4 (b128) gather + 4 fp32 atomics.
__global__ void gcn_scatter128(const float* __restrict__ H1,
                               const int* __restrict__ src, const int* __restrict__ dst,
                               float* __restrict__ AGG, int n_edges) {
  const int e    = blockIdx.x * (blockDim.x >> 5) + (threadIdx.x >> 5);
  const int lane = threadIdx.x & 31;
  if (e >= n_edges) return;
  const int s = src[e];
  const int d = dst[e];
  const float4 v = ((const float4*)(H1 + (size_t)s * GCN_HID))[lane];
  float* p = AGG + (size_t)d * GCN_HID + lane * 4;
  atomAddF(p + 0, v.x);
  atomAddF(p + 1, v.y);
  atomAddF(p + 2, v.z);
  atomAddF(p + 3, v.w);
}

// ---- layer-2 GEMM: H2 = (relu(AGG1*norm_dst + b1) @ W2) * norm_src --------
// Epilogue of layer 1 (dst-norm, bias, relu) fused into the A-operand load,
// saving a full 25.6 MB read+write pass. W2S is pre-padded so no N guards in
// the compute loop.
__global__ void __launch_bounds__(32) gcn_gemm2_wmma(
    const float* __restrict__ AGG, const float* __restrict__ b1,
    const float* __restrict__ W2S, const float* __restrict__ norm_src,
    const float* __restrict__ norm_dst, float* __restrict__ H2) {
  const int lane = threadIdx.x & 31;
  const int lm   = lane & 15;
  const int kh   = lane >> 4;
  const int m0   = blockIdx.x * 16;
  const int row  = m0 + lm;
  const float nd = norm_dst[row];

  v8f acc[3] = {};
  const float* arow = AGG + (size_t)row * GCN_HID;
  const v2f*   Bv   = (const v2f*)W2S;

#pragma unroll 2
  for (int s = 0; s < GCN_KS2; ++s) {
    const int kk = 4 * s + kh * 2;
    v2f raw = *(const v2f*)(arow + kk);
    v2f a;
    a.x = fmaxf(raw.x * nd + b1[kk + 0], 0.0f);     // fused bias + relu
    a.y = fmaxf(raw.y * nd + b1[kk + 1], 0.0f);
    v2f bf[3];
#pragma unroll
    for (int nt = 0; nt < 3; ++nt)
      bf[nt] = Bv[(s * 3 + nt) * 32 + lane];
#pragma unroll
    for (int nt = 0; nt < 3; ++nt)
      acc[nt] = __builtin_amdgcn_wmma_f32_16x16x4_f32(
          false, a, false, bf[nt], (short)0, acc[nt], false, false);
  }

#pragma unroll
  for (int r = 0; r < 8; ++r) {
    const int mrow = m0 + kh * 8 + r;
    const float ns = norm_src[mrow];
#pragma unroll
    for (int nt = 0; nt < 3; ++nt) {
      const int n = nt * 16 + lm;
      if (n < GCN_CLS)
        H2[(size_t)mrow * GCN_CLS + n] = acc[nt][r] * ns;
    }
  }
}

// ---- layer-2 scatter: OUT[dst] += H2[src], 40 feats (flat edge*feat) ------
__global__ void gcn_scatter40(const float* __restrict__ H2,
                              const int* __restrict__ src, const int* __restrict__ dst,
                              float* __restrict__ OUT, int n_edges) {
  int idx = blockIdx.x * blockDim.x + threadIdx.x;
  int tot = n_edges * GCN_CLS;               // 64M < 2^31
  if (idx >= tot) return;
  int e = idx / GCN_CLS;
  int f = idx - e * GCN_CLS;
  atomAddF(&OUT[(size_t)dst[e] * GCN_CLS + f], H2[(size_t)src[e] * GCN_CLS + f]);
}

// ---- finalize: OUT = OUT * norm_dst[row] + b2[col] ------------------------
__global__ void gcn_finalize(float* __restrict__ OUT, const float* __restrict__ norm_dst,
                             const float* __restrict__ b2, int n_nodes) {
  int idx = blockIdx.x * blockDim.x + threadIdx.x;
  int tot = n_nodes * GCN_CLS;
  if (idx >= tot) return;
  int i = idx / GCN_CLS;
  int f = idx - i * GCN_CLS;
  OUT[idx] = OUT[idx] * norm_dst[i] + b2[f];
}

// ---------------------------------------------------------------------------
extern "C" void kernel_launch(void* const* d_in, const int* in_sizes, int n_in,
                              void* d_out, int out_size, void* d_ws, size_t ws_size,
                              hipStream_t stream) {
  const float* X   = (const float*)d_in[0];
  const float* W1  = (const float*)d_in[1];
  const float* b1  = (const float*)d_in[2];
  const float* W2  = (const float*)d_in[3];
  const float* b2  = (const float*)d_in[4];
  const int*   src = (const int*)d_in[5];
  const int*   dst = (const int*)d_in[6];
  const int n_edges = in_sizes[5];
  const int n_nodes = GCN_NODES;

  // workspace layout (256B-aligned slabs), ~60 MB total
  char* ws = (char*)d_ws;
  size_t off = 0;
  auto slab = [&](size_t bytes) -> void* {
    void* p = ws + off;
    off += (bytes + 255) & ~(size_t)255;
    return p;
  };
  float* deg_out  = (float*)slab((size_t)n_nodes * 4);
  float* deg_in   = (float*)slab((size_t)n_nodes * 4);
  float* norm_src = (float*)slab((size_t)n_nodes * 4);
  float* norm_dst = (float*)slab((size_t)n_nodes * 4);
  float* W1S      = (float*)slab((size_t)GCN_KS1 * 8 * 32 * 2 * 4);  // 256 KB
  float* W2S      = (float*)slab((size_t)GCN_KS2 * 3 * 32 * 2 * 4);  // 24 KB
  float* H1       = (float*)slab((size_t)n_nodes * GCN_HID * 4);
  float* AGG1     = (float*)slab((size_t)n_nodes * GCN_HID * 4);
  float* H2       = (float*)slab((size_t)n_nodes * GCN_CLS * 4);
  (void)ws_size;

  float* OUT = (float*)d_out;

  // 1) degrees + norms + weight pre-swizzle
  hipMemsetAsync(deg_out, 0, (size_t)n_nodes * 4, stream);
  hipMemsetAsync(deg_in, 0, (size_t)n_nodes * 4, stream);
  gcn_deg_kernel<<<(n_edges + 255) / 256, 256, 0, stream>>>(src, dst, deg_out, deg_in, n_edges);
  gcn_norm_kernel<<<(n_nodes + 255) / 256, 256, 0, stream>>>(deg_out, deg_in, norm_src, norm_dst, n_nodes);
  gcn_pack_w1<<<(GCN_KS1 * 8 * 32 + 255) / 256, 256, 0, stream>>>(W1, W1S);
  gcn_pack_w2<<<(GCN_KS2 * 3 * 32 + 255) / 256, 256, 0, stream>>>(W2, W2S);

  // 2) layer-1 GEMM (WMMA f32), norm_src fused into store
  gcn_gemm1_wmma<<<n_nodes / 16, 32, 0, stream>>>(X, W1S, norm_src, H1);

  // 3) layer-1 aggregation
  hipMemsetAsync(AGG1, 0, (size_t)n_nodes * GCN_HID * 4, stream);
  gcn_scatter128<<<(n_edges + 7) / 8, 256, 0, stream>>>(H1, src, dst, AGG1, n_edges);

  // 4) layer-2 GEMM (WMMA f32) with fused dst-norm + bias + relu on input
  gcn_gemm2_wmma<<<n_nodes / 16, 32, 0, stream>>>(AGG1, b1, W2S, norm_src, norm_dst, H2);

  // 5) layer-2 aggregation into d_out, then finalize (dst-norm + bias)
  hipMemsetAsync(OUT, 0, (size_t)n_nodes * GCN_CLS * 4, stream);
  {
    int tot = n_edges * GCN_CLS;
    gcn_scatter40<<<(tot + 255) / 256, 256, 0, stream>>>(H2, src, dst, OUT, n_edges);
  }
  {
    int tot = n_nodes * GCN_CLS;
    gcn_finalize<<<(tot + 255) / 256, 256, 0, stream>>>(OUT, norm_dst, b2, n_nodes);
  }
}